// GCN_34248069218824
// MI455X (gfx1250) — compile-verified
//
#include <hip/hip_runtime.h>
#include <hip/hip_bf16.h>
#include <math.h>

#define B_GRAPHS 4096
#define N_NODES  128
#define E_PER    1024
#define F_DIM    19
#define E_TOTAL  (B_GRAPHS * E_PER)

typedef float v2f __attribute__((ext_vector_type(2)));
typedef float v8f __attribute__((ext_vector_type(8)));

// ---------------------------------------------------------------------------
// Utility: zero int buffer (kept-mask must be re-zeroed every call)
// ---------------------------------------------------------------------------
__global__ void zero_i32_kernel(int* __restrict__ p, int n) {
    int i = blockIdx.x * blockDim.x + threadIdx.x;
    if (i < n) p[i] = 0;
}

// ---------------------------------------------------------------------------
// H = X @ W^T  with W [19,19], X [num_nodes,19], via V_WMMA_F32_16X16X4_F32.
// Each wave computes a 16-row tile; K padded 19->20 (5 steps of 4), N padded
// 19->32 (two 16-wide WMMA tiles).  W^T staged zero-padded in LDS.
// ---------------------------------------------------------------------------
__global__ void linear19_wmma_kernel(const float* __restrict__ X,
                                     const float* __restrict__ W,
                                     float* __restrict__ H, int num_nodes) {
    __shared__ float Wt[20][32];           // Wt[k][n] = W[n][k], zero padded
    const int tid = threadIdx.x;
    for (int i = tid; i < 20 * 32; i += blockDim.x) {
        int k = i >> 5, n = i & 31;
        Wt[k][n] = (k < F_DIM && n < F_DIM) ? W[n * F_DIM + k] : 0.0f;
    }
    __syncthreads();

    const int wave = tid >> 5;
    const int lane = tid & 31;
    const int half = lane >> 4;            // 0: lanes 0-15, 1: lanes 16-31
    const int l    = lane & 15;
    const int base = (blockIdx.x * 8 + wave) * 16;
    if (base >= num_nodes) return;         // uniform per wave -> EXEC stays full

    v8f c0 = {}; v8f c1 = {};
    const int rowM = base + l;             // A: M striped across lanes (both halves)

#pragma unroll
    for (int kk = 0; kk < 5; ++kk) {
        const int k0 = kk * 4 + half * 2;  // K across {VGPR, lane-half}
        const int k1 = k0 + 1;
        v2f a;
        a.x = (k0 < F_DIM) ? X[(size_t)rowM * F_DIM + k0] : 0.0f;
        a.y = (k1 < F_DIM) ? X[(size_t)rowM * F_DIM + k1] : 0.0f;
        v2f b0 = { Wt[k0][l],      Wt[k1][l]      };
        v2f b1 = { Wt[k0][l + 16], Wt[k1][l + 16] };
#if defined(__HIP_DEVICE_COMPILE__)
        c0 = __builtin_amdgcn_wmma_f32_16x16x4_f32(false, a, false, b0, (short)0, c0, false, false);
        c1 = __builtin_amdgcn_wmma_f32_16x16x4_f32(false, a, false, b1, (short)0, c1, false, false);
#else
        (void)a; (void)b0; (void)b1;
#endif
    }

#pragma unroll
    for (int v = 0; v < 8; ++v) {          // D: VGPR v -> M = v + half*8, N = lane%16
        const int m   = v + half * 8;
        const int row = base + m;
        if (l < F_DIM)      H[(size_t)row * F_DIM + l]      = c0[v];
        if (l + 16 < F_DIM) H[(size_t)row * F_DIM + l + 16] = c1[v];
    }
}

// ---------------------------------------------------------------------------
// Fused GCNConv aggregation: one block per graph.  Stages the graph's edges,
// feature tile and deg^-1/2 in LDS; deterministic (no atomics).
//   out_i = relu( sum_j dinv_i*dinv_j*h_j  +  dinv_i^2 * h_i  +  b )
// ---------------------------------------------------------------------------
__global__ void gcn_fused_kernel(const float* __restrict__ H,
                                 const int* __restrict__ src,
                                 const int* __restrict__ dst,
                                 const int* __restrict__ valid,   // NULL => all valid
                                 const float* __restrict__ bias,
                                 float* __restrict__ out, int npg) {
    __shared__ int   ls[E_PER];            // local src (-1 = invalid edge)
    __shared__ int   ld[E_PER];            // local dst (-1 = invalid edge)
    __shared__ float sh[N_NODES * F_DIM];  // graph feature tile (max 128x19)
    __shared__ float sdinv[N_NODES];

    const int g = blockIdx.x;
    const int t = threadIdx.x;
    const size_t ebase = (size_t)g * E_PER;
    const size_t nbase = (size_t)g * npg;

    for (int i = t; i < E_PER; i += blockDim.x) {
        int v = valid ? valid[ebase + i] : 1;
        ls[i] = v ? (src[ebase + i] - (int)nbase) : -1;
        ld[i] = v ? (dst[ebase + i] - (int)nbase) : -1;
    }
    for (int i = t; i < npg * F_DIM; i += blockDim.x)
        sh[i] = H[nbase * F_DIM + i];
    __syncthreads();

    if (t < npg) {                         // degree (valid in-edges) + self loop
        int cnt = 0;
        for (int i = 0; i < E_PER; ++i) cnt += (ld[i] == t);
        sdinv[t] = rsqrtf((float)cnt + 1.0f);
    }
    __syncthreads();

    if (t < npg) {
        float acc[F_DIM];
#pragma unroll
        for (int f = 0; f < F_DIM; ++f) acc[f] = 0.0f;
        const float di = sdinv[t];
        for (int i = 0; i < E_PER; ++i) {
            if (ld[i] == t) {
                const int   s    = ls[i];
                const float coef = sdinv[s] * di;
#pragma unroll
                for (int f = 0; f < F_DIM; ++f)
                    acc[f] = fmaf(sh[s * F_DIM + f], coef, acc[f]);
            }
        }
        const float self = di * di;
#pragma unroll
        for (int f = 0; f < F_DIM; ++f) {
            float v = acc[f] + sh[t * F_DIM + f] * self + bias[f];
            out[(nbase + t) * F_DIM + f] = fmaxf(v, 0.0f);
        }
    }
}

// ---------------------------------------------------------------------------
// Per-graph feature-sum pool (nodes contiguous per graph).
// ---------------------------------------------------------------------------
__global__ void graph_pool_kernel(const float* __restrict__ X,
                                  float* __restrict__ pool, int npg) {
    const int g = blockIdx.x;
    const int f = threadIdx.x;
    if (f >= F_DIM) return;
    float s = 0.0f;
    const size_t nbase = (size_t)g * npg;
    for (int i = 0; i < npg; ++i) s += X[(nbase + i) * F_DIM + f];
    pool[(size_t)g * F_DIM + f] = s;
}

// ---------------------------------------------------------------------------
// TopKPooling(ratio=0.5, tanh): score = x.w/||w||, keep top n/2 per graph via
// in-LDS bitonic sort (descending), scale kept rows by tanh(score), emit
// kept-mask + old->new index map.  Block = n threads (128 or 64).
// ---------------------------------------------------------------------------
__global__ void topk_kernel(const float* __restrict__ Xin,
                            const float* __restrict__ w,
                            float* __restrict__ Xout,
                            int* __restrict__ kept,
                            int* __restrict__ invmap, int n) {
    __shared__ float sc[N_NODES];
    __shared__ int   si[N_NODES];
    const int g = blockIdx.x;
    const int t = threadIdx.x;

    float wn = 0.0f;
#pragma unroll
    for (int f = 0; f < F_DIM; ++f) wn += w[f] * w[f];
    wn = rsqrtf(wn);

    const size_t node = (size_t)g * n + t;
    float s = 0.0f;
#pragma unroll
    for (int f = 0; f < F_DIM; ++f) s += Xin[node * F_DIM + f] * w[f];
    sc[t] = s * wn;
    si[t] = t;

    for (int ksz = 2; ksz <= n; ksz <<= 1) {
        for (int j = ksz >> 1; j > 0; j >>= 1) {
            __syncthreads();
            const int ixj = t ^ j;
            if (ixj > t) {
                const bool  up = ((t & ksz) == 0);     // descending segments
                const float a = sc[t], b = sc[ixj];
                if (up ? (a < b) : (a > b)) {
                    sc[t] = b; sc[ixj] = a;
                    int tmp = si[t]; si[t] = si[ixj]; si[ixj] = tmp;
                }
            }
        }
    }
    __syncthreads();

    const int k = n >> 1;
    if (t < k) {
        const int    oldg  = g * n + si[t];
        const int    newg  = g * k + t;
        const float  scale = tanhf(sc[t]);
#pragma unroll
        for (int f = 0; f < F_DIM; ++f)
            Xout[(size_t)newg * F_DIM + f] = Xin[(size_t)oldg * F_DIM + f] * scale;
        kept[oldg]   = 1;
        invmap[oldg] = newg;
    }
}

// ---------------------------------------------------------------------------
// Remap edges through the kept/inv maps.
// ---------------------------------------------------------------------------
__global__ void edge_remap_kernel(const int* __restrict__ src_in,
                                  const int* __restrict__ dst_in,
                                  const int* __restrict__ valid_in,  // NULL => all
                                  const int* __restrict__ kept,
                                  const int* __restrict__ invmap,
                                  int* __restrict__ src_out,
                                  int* __restrict__ dst_out,
                                  int* __restrict__ valid_out, int nE) {
    int e = blockIdx.x * blockDim.x + threadIdx.x;
    if (e >= nE) return;
    __builtin_prefetch(&src_in[e + 4096], 0, 1);   // global_prefetch_b8
    const int s = src_in[e];
    const int d = dst_in[e];
    const int v = (valid_in ? valid_in[e] : 1) && kept[s] && kept[d];
    valid_out[e] = v;
    src_out[e]   = v ? invmap[s] : 0;
    dst_out[e]   = v ? invmap[d] : 0;
}

// ---------------------------------------------------------------------------
// MLP layer 1: z32 = relu(concat(x1,x2,x3) @ L1W^T + L1b)  [4096,57]->[4096,32]
// WMMA f32 16x16x4, K padded 57->60, N = 32 (two tiles).
// ---------------------------------------------------------------------------
__device__ inline float concat_fetch(const float* x1, const float* x2,
                                     const float* x3, int row, int k) {
    if (k < 19) return x1[(size_t)row * F_DIM + k];
    if (k < 38) return x2[(size_t)row * F_DIM + (k - 19)];
    if (k < 57) return x3[(size_t)row * F_DIM + (k - 38)];
    return 0.0f;
}

__global__ void mlp1_wmma_kernel(const float* __restrict__ x1,
                                 const float* __restrict__ x2,
                                 const float* __restrict__ x3,
                                 const float* __restrict__ L1W,
                                 const float* __restrict__ L1b,
                                 float* __restrict__ z32) {
    __shared__ float Wt[60][32];           // Wt[k][n] = L1W[n][k], zero padded
    const int tid = threadIdx.x;
    for (int i = tid; i < 60 * 32; i += blockDim.x) {
        int k = i >> 5, n = i & 31;
        Wt[k][n] = (k < 57) ? L1W[n * 57 + k] : 0.0f;
    }
    __syncthreads();

    const int wave = tid >> 5;
    const int lane = tid & 31;
    const int half = lane >> 4;
    const int l    = lane & 15;
    const int base = (blockIdx.x * 8 + wave) * 16;

    v8f c0 = {}; v8f c1 = {};
    const int rowM = base + l;

    for (int kk = 0; kk < 15; ++kk) {
        const int k0 = kk * 4 + half * 2;
        const int k1 = k0 + 1;
        v2f a;
        a.x = concat_fetch(x1, x2, x3, rowM, k0);
        a.y = concat_fetch(x1, x2, x3, rowM, k1);
        v2f b0 = { Wt[k0][l],      Wt[k1][l]      };
        v2f b1 = { Wt[k0][l + 16], Wt[k1][l + 16] };
#if defined(__HIP_DEVICE_COMPILE__)
        c0 = __builtin_amdgcn_wmma_f32_16x16x4_f32(false, a, false, b0, (short)0, c0, false, false);
        c1 = __builtin_amdgcn_wmma_f32_16x16x4_f32(false, a, false, b1, (short)0, c1, false, false);
#else
        (void)a; (void)b0; (void)b1;
#endif
    }

#pragma unroll
    for (int v = 0; v < 8; ++v) {
        const int m   = v + half * 8;
        const int row = base + m;
        z32[(size_t)row * 32 + l]      = fmaxf(c0[v] + L1b[l],      0.0f);
        z32[(size_t)row * 32 + l + 16] = fmaxf(c1[v] + L1b[l + 16], 0.0f);
    }
}

// ---------------------------------------------------------------------------
// MLP tail: 32 -> relu(10) -> 1 -> sigmoid.  One thread per row.
// ---------------------------------------------------------------------------
__global__ void mlp_tail_kernel(const float* __restrict__ z32,
                                const float* __restrict__ L2W,
                                const float* __restrict__ L2b,
                                const float* __restrict__ L3W,
                                const float* __restrict__ L3b,
                                float* __restrict__ out, int nrows) {
    int r = blockIdx.x * blockDim.x + threadIdx.x;
    if (r >= nrows) return;
    float z[32];
#pragma unroll
    for (int i = 0; i < 32; ++i) z[i] = z32[(size_t)r * 32 + i];
    float y = L3b[0];
#pragma unroll
    for (int j = 0; j < 10; ++j) {
        float h = L2b[j];
#pragma unroll
        for (int i = 0; i < 32; ++i) h = fmaf(z[i], L2W[j * 32 + i], h);
        y = fmaf(fmaxf(h, 0.0f), L3W[j], y);
    }
    out[r] = 1.0f / (1.0f + expf(-y));
}

// ---------------------------------------------------------------------------
// Host launcher
// ---------------------------------------------------------------------------
extern "C" void kernel_launch(void* const* d_in, const int* in_sizes, int n_in,
                              void* d_out, int out_size, void* d_ws, size_t ws_size,
                              hipStream_t stream) {
    (void)in_sizes; (void)n_in; (void)out_size; (void)ws_size;

    const float* x      = (const float*)d_in[0];
    const int*   src0   = (const int*)  d_in[1];
    const int*   dst0   = (const int*)  d_in[2];
    /* d_in[3] = batch (unused: nodes are contiguous per graph) */
    const float* W1     = (const float*)d_in[4];
    const float* b1     = (const float*)d_in[5];
    const float* W2     = (const float*)d_in[6];
    const float* b2     = (const float*)d_in[7];
    const float* W3     = (const float*)d_in[8];
    const float* b3     = (const float*)d_in[9];
    const float* topk_w = (const float*)d_in[10];
    const float* L1W    = (const float*)d_in[11];
    const float* L1b    = (const float*)d_in[12];
    const float* L2W    = (const float*)d_in[13];
    const float* L2b    = (const float*)d_in[14];
    const float* L3W    = (const float*)d_in[15];
    const float* L3b    = (const float*)d_in[16];
    float* out = (float*)d_out;

    const size_t NODES1 = (size_t)B_GRAPHS * N_NODES;   // 524288
    const size_t NODES2 = NODES1 / 2;                   // 262144
    const size_t NODES3 = NODES1 / 4;                   // 131072

    // Bump allocator over d_ws (256B aligned chunks).
    unsigned char* ws = (unsigned char*)d_ws;
    size_t off = 0;
    auto take = [&](size_t bytes) -> void* {
        void* p = ws + off;
        off = (off + bytes + 255) & ~(size_t)255;
        return p;
    };
    float* d_h    = (float*)take(NODES1 * F_DIM * sizeof(float));
    float* d_xa   = (float*)take(NODES1 * F_DIM * sizeof(float));
    float* d_xb   = (float*)take(NODES2 * F_DIM * sizeof(float));
    int*   d_kept = (int*)  take(NODES1 * sizeof(int));
    int*   d_inv  = (int*)  take(NODES1 * sizeof(int));
    int*   eA_src = (int*)  take((size_t)E_TOTAL * sizeof(int));
    int*   eA_dst = (int*)  take((size_t)E_TOTAL * sizeof(int));
    int*   eA_val = (int*)  take((size_t)E_TOTAL * sizeof(int));
    int*   eB_src = (int*)  take((size_t)E_TOTAL * sizeof(int));
    int*   eB_dst = (int*)  take((size_t)E_TOTAL * sizeof(int));
    int*   eB_val = (int*)  take((size_t)E_TOTAL * sizeof(int));
    float* pool1  = (float*)take((size_t)B_GRAPHS * F_DIM * sizeof(float));
    float* pool2  = (float*)take((size_t)B_GRAPHS * F_DIM * sizeof(float));
    float* pool3  = (float*)take((size_t)B_GRAPHS * F_DIM * sizeof(float));
    float* d_z32  = (float*)take((size_t)B_GRAPHS * 32 * sizeof(float));

    const int eBlocks = (E_TOTAL + 255) / 256;

    // ---- Layer 1 (full graph, 128 nodes/graph) ----
    linear19_wmma_kernel<<<(int)(NODES1 / 128), 256, 0, stream>>>(x, W1, d_h, (int)NODES1);
    gcn_fused_kernel<<<B_GRAPHS, 128, 0, stream>>>(d_h, src0, dst0, nullptr, b1, d_xa, N_NODES);
    graph_pool_kernel<<<B_GRAPHS, 32, 0, stream>>>(d_xa, pool1, N_NODES);

    zero_i32_kernel<<<(int)((NODES1 + 255) / 256), 256, 0, stream>>>(d_kept, (int)NODES1);
    topk_kernel<<<B_GRAPHS, 128, 0, stream>>>(d_xa, topk_w, d_xb, d_kept, d_inv, N_NODES);
    edge_remap_kernel<<<eBlocks, 256, 0, stream>>>(src0, dst0, nullptr, d_kept, d_inv,
                                                   eA_src, eA_dst, eA_val, E_TOTAL);

    // ---- Layer 2 (64 nodes/graph) ----
    linear19_wmma_kernel<<<(int)(NODES2 / 128), 256, 0, stream>>>(d_xb, W2, d_h, (int)NODES2);
    gcn_fused_kernel<<<B_GRAPHS, 128, 0, stream>>>(d_h, eA_src, eA_dst, eA_val, b2, d_xa, N_NODES / 2);
    graph_pool_kernel<<<B_GRAPHS, 32, 0, stream>>>(d_xa, pool2, N_NODES / 2);

    zero_i32_kernel<<<(int)((NODES2 + 255) / 256), 256, 0, stream>>>(d_kept, (int)NODES2);
    topk_kernel<<<B_GRAPHS, 64, 0, stream>>>(d_xa, topk_w, d_xb, d_kept, d_inv, N_NODES / 2);
    edge_remap_kernel<<<eBlocks, 256, 0, stream>>>(eA_src, eA_dst, eA_val, d_kept, d_inv,
                                                   eB_src, eB_dst, eB_val, E_TOTAL);

    // ---- Layer 3 (32 nodes/graph) ----
    linear19_wmma_kernel<<<(int)(NODES3 / 128), 256, 0, stream>>>(d_xb, W3, d_h, (int)NODES3);
    gcn_fused_kernel<<<B_GRAPHS, 128, 0, stream>>>(d_h, eB_src, eB_dst, eB_val, b3, d_xa, N_NODES / 4);
    graph_pool_kernel<<<B_GRAPHS, 32, 0, stream>>>(d_xa, pool3, N_NODES / 4);

    // ---- MLP head ----
    mlp1_wmma_kernel<<<B_GRAPHS / 128, 256, 0, stream>>>(pool1, pool2, pool3, L1W, L1b, d_z32);
    mlp_tail_kernel<<<B_GRAPHS / 256, 256, 0, stream>>>(d_z32, L2W, L2b, L3W, L3b, out, B_GRAPHS);
}